// YOLOLoss_27779848471398
// MI455X (gfx1250) — compile-verified
//
#include <hip/hip_runtime.h>
#include <math.h>

#define NCLS   80
#define NCH    85      // NCLS + 5
#define BATCH  32
#define MAXT   50
#define NA     3

// per-scale geometry
#define N0 16224       // 32*3*13*13
#define N1 64896       // 32*3*26*26
#define N2 259584      // 32*3*52*52
#define TOTAL (N0 + N1 + N2)

__constant__ float d_anchors[3][3][2] = {
    {{0.28f, 0.22f}, {0.38f, 0.48f}, {0.90f, 0.78f}},
    {{0.07f, 0.15f}, {0.15f, 0.11f}, {0.14f, 0.29f}},
    {{0.02f, 0.03f}, {0.04f, 0.07f}, {0.08f, 0.06f}},
};

typedef __attribute__((ext_vector_type(2))) float v2f;
typedef __attribute__((ext_vector_type(8))) float v8f;

// Full-wave (32 lane) sum using V_WMMA_F32_16X16X4_F32.
// A (16x4) packed as {acc, 0} per lane => A[m,0]=acc[m], A[m,2]=acc[m+16].
// B = ones (4x16) => D[m][n] = acc[m] + acc[m+16] (row sums).
// Lane n<16 holds D rows 0..7 in its 8 VGPRs; n>=16 holds rows 8..15.
// Sum the 8 VGPRs, then fold halves with one xor-shuffle: total in every lane.
// Caller must be at a wave-uniform point (EXEC all ones).
__device__ __forceinline__ float wave_reduce_wmma(float acc) {
    v2f a; a[0] = acc;  a[1] = 0.0f;
    v2f b; b[0] = 1.0f; b[1] = 1.0f;
    v8f c = {};
    c = __builtin_amdgcn_wmma_f32_16x16x4_f32(
        /*neg_a=*/false, a, /*neg_b=*/false, b,
        /*c_mod=*/(short)0, c, /*reuse_a=*/false, /*reuse_b=*/false);
    float r = c[0] + c[1] + c[2] + c[3] + c[4] + c[5] + c[6] + c[7];
    r += __shfl_xor(r, 16, 32);
    return r;
}

__device__ __forceinline__ float bce_pos(float x) {     // bce(x, t=0 part): max(x,0)+log1p(exp(-|x|))
    return fmaxf(x, 0.0f) + log1pf(expf(-fabsf(x)));
}

__device__ __forceinline__ unsigned pcg_hash(unsigned v) {
    v = v * 747796405u + 2891336453u;
    unsigned w = ((v >> ((v >> 28) + 4u)) ^ v) * 277803737u;
    return (w >> 22) ^ w;
}

// -------------------- kernel 1: init workspace --------------------
__global__ void __launch_bounds__(256)
k_init(int* __restrict__ objmap, float* __restrict__ fws) {
    int idx = blockIdx.x * blockDim.x + threadIdx.x;
    if (idx < TOTAL) objmap[idx] = -1;
    if (idx < 16)    fws[idx] = 0.0f;
}

// -------------------- kernel 2: target assignment (scatter) --------------------
__global__ void __launch_bounds__(256)
k_assign(const float* __restrict__ targets, int* __restrict__ objmap,
         float* __restrict__ fws) {
    int tid = blockIdx.x * blockDim.x + threadIdx.x;
    if (tid >= 3 * BATCH * MAXT) return;
    int s = tid / (BATCH * MAXT);
    int r = tid - s * (BATCH * MAXT);
    int b = r / MAXT;
    int t = r - b * MAXT;

    const int Gs[3]   = {13, 26, 52};
    const int offs[3] = {0, N0, N0 + N1};
    int G = Gs[s];

    const float* tg = targets + (b * MAXT + t) * 5;
    float clsf = tg[0];
    int cls = (int)clsf;                       // trunc-toward-zero, matches astype(int32)
    if (!(cls >= 0 && cls < NCLS)) return;

    float gx = tg[1] * (float)G, gy = tg[2] * (float)G;
    int gi = (int)gx, gj = (int)gy;
    if (!(gi >= 0 && gi < G && gj >= 0 && gj < G)) return;

    float twn = tg[3], thn = tg[4];
    int best = 0; float biou = -1.0f;
#pragma unroll
    for (int a = 0; a < NA; ++a) {
        float aw = d_anchors[s][a][0], ah = d_anchors[s][a][1];
        float inter = fmaxf(fminf(twn, aw), 0.0f) * fmaxf(fminf(thn, ah), 0.0f);
        float iou = inter / (aw * ah + twn * thn - inter + 1e-16f);
        if (iou > biou) { biou = iou; best = a; }   // strict '>' keeps first max (argmax semantics)
    }
    if (!(biou > 0.2f)) return;

    int flat = ((b * NA + best) * G + gj) * G + gi;
    int old = atomicMax(&objmap[offs[s] + flat], t);  // deterministic winner (max t)
    if (old < 0) atomicAdd(&fws[s * 5 + 0], 1.0f);    // exact distinct-cell count
}

// -------------------- kernel 3: dense per-scale pass (all loss terms) --------------------
__global__ void __launch_bounds__(256)
k_main(const float* __restrict__ pred, const float* __restrict__ targets,
       const int* __restrict__ objmap_s, float* __restrict__ fws,
       int s, int G, int N) {
    int idx = blockIdx.x * blockDim.x + threadIdx.x;
    float a_coord = 0.0f, a_obj = 0.0f, a_cls = 0.0f, a_noobj = 0.0f;

    if (idx < N) {
        int GG = G * G;
        int gi = idx % G;
        int t2 = idx / G;
        int gj = t2 % G;
        int t3 = t2 / G;
        int a  = t3 % NA;
        int b  = t3 / NA;
        int base = ((b * (NA * NCH) + a * NCH) * G + gj) * G + gi;

        int t = objmap_s[idx];
        if (t >= 0) {
            // ---- object cell: coord + obj + cls terms ----
            const float* tg = targets + (b * MAXT + t) * 5;
            int   cls = (int)tg[0];
            float gx = tg[1] * (float)G, gy = tg[2] * (float)G;
            float gw = tg[3] * (float)G, gh = tg[4] * (float)G;
            float tx = gx - (float)gi;
            float ty = gy - (float)gj;
            float aw = d_anchors[s][a][0], ah = d_anchors[s][a][1];
            float tw = logf(gw / aw + 1e-16f);
            float th = logf(gh / ah + 1e-16f);

            float x0 = pred[base];
            float x1 = pred[base + GG];
            float x2 = pred[base + 2 * GG];
            float x3 = pred[base + 3 * GG];
            float x4 = pred[base + 4 * GG];
            float s0 = 1.0f / (1.0f + expf(-x0));
            float s1 = 1.0f / (1.0f + expf(-x1));
            a_coord = (s0 - tx) * (s0 - tx) + (s1 - ty) * (s1 - ty)
                    + (x2 - tw) * (x2 - tw) + (x3 - th) * (x3 - th);
            a_obj = bce_pos(x4) - x4;   // bce(x4, t=1) = max(x,0)-x+log1p(exp(-|x|))

            float cl = 0.0f;
            for (int c = 0; c < NCLS; ++c) {
                float xc = pred[base + (5 + c) * GG];
                cl += bce_pos(xc);
                if (c == cls) cl -= xc; // -x*t, t one-hot
            }
            a_cls = cl;
        } else {
            // ---- noobj cell: random-subset BCE(x4, 0) ----
            float cnt   = fws[s * 5 + 0];        // obj_count (written by k_assign)
            int objc    = (int)cnt;
            int noobjc  = N - objc;
            int maxno   = 3 * objc; if (maxno > noobjc) maxno = noobjc;
            if (maxno > 0) {
                float prob = (float)maxno / (float)noobjc;
                unsigned h = pcg_hash((unsigned)idx * 9781u + (unsigned)s * 0x9E3779B9u);
                if ((float)h * 2.3283064e-10f < prob) {
                    float x4 = pred[base + 4 * GG];
                    a_noobj = bce_pos(x4);
                }
            }
        }
    }

    // wave-uniform reduction point: EXEC is all-ones here (tail lanes contributed 0)
    float r_coord = wave_reduce_wmma(a_coord);
    float r_obj   = wave_reduce_wmma(a_obj);
    float r_cls   = wave_reduce_wmma(a_cls);
    float r_noobj = wave_reduce_wmma(a_noobj);
    if ((threadIdx.x & 31) == 0) {
        atomicAdd(&fws[s * 5 + 1], r_coord);
        atomicAdd(&fws[s * 5 + 2], r_obj);
        atomicAdd(&fws[s * 5 + 3], r_cls);
        atomicAdd(&fws[s * 5 + 4], r_noobj);
    }
}

// -------------------- kernel 4: final combine --------------------
__global__ void k_final(const float* __restrict__ fws, float* __restrict__ out) {
    if (threadIdx.x != 0 || blockIdx.x != 0) return;
    const int Ns[3] = {N0, N1, N2};
    float coord = 0.f, obj = 0.f, noobj = 0.f, cls = 0.f, num = 0.f;
    for (int s = 0; s < 3; ++s) {
        float cnt = fws[s * 5 + 0];
        if (cnt > 0.0f) {                      // 'active' gate
            float safe = fmaxf(cnt, 1.0f);
            coord += fws[s * 5 + 1] / (2.0f * safe);
            obj   += fws[s * 5 + 2] / safe;
            cls   += fws[s * 5 + 3] / (safe * (float)NCLS);
            int objc = (int)cnt;
            int noobjc = Ns[s] - objc;
            int maxno = 3 * objc; if (maxno > noobjc) maxno = noobjc;
            noobj += fws[s * 5 + 4] / fmaxf((float)maxno, 1.0f);
            num += 1.0f;
        }
    }
    float total = (5.0f * coord + obj + 0.5f * noobj + cls) / fmaxf(num, 1.0f);
    out[0] = (num > 0.0f) ? total : 0.1f;
}

// -------------------- host launcher --------------------
extern "C" void kernel_launch(void* const* d_in, const int* in_sizes, int n_in,
                              void* d_out, int out_size, void* d_ws, size_t ws_size,
                              hipStream_t stream) {
    const float* p0 = (const float*)d_in[0];
    const float* p1 = (const float*)d_in[1];
    const float* p2 = (const float*)d_in[2];
    const float* tg = (const float*)d_in[3];

    int*   objmap = (int*)d_ws;
    float* fws    = (float*)((char*)d_ws + (size_t)TOTAL * sizeof(int));
    float* out    = (float*)d_out;

    k_init<<<(TOTAL + 255) / 256, 256, 0, stream>>>(objmap, fws);
    k_assign<<<(3 * BATCH * MAXT + 255) / 256, 256, 0, stream>>>(tg, objmap, fws);

    const float* preds[3] = {p0, p1, p2};
    const int Gs[3]   = {13, 26, 52};
    const int Ns[3]   = {N0, N1, N2};
    const int offs[3] = {0, N0, N0 + N1};
    for (int s = 0; s < 3; ++s) {
        k_main<<<(Ns[s] + 255) / 256, 256, 0, stream>>>(
            preds[s], tg, objmap + offs[s], fws, s, Gs[s], Ns[s]);
    }
    k_final<<<1, 32, 0, stream>>>(fws, out);
}